// DetectionLoss_6287832121366
// MI455X (gfx1250) — compile-verified
//
#include <hip/hip_runtime.h>
#include <hip/hip_bf16.h>
#include <stdint.h>

#ifndef __has_builtin
#define __has_builtin(x) 0
#endif

// Problem constants (match reference setup_inputs)
#define BDIM  16
#define NPRED 8400
#define NTGT  30
#define NCLS  80
#define PDIM  85   // 5 + NUM_CLASSES
#define TDIM  5
#define EPSF  1e-7f
#define MBS   1024 // match-kernel block size (32 wave32 waves)

// ---------------- dynamic LDS layout for the matching kernel ----------------
#define NP1 (NPRED + 1)
static constexpr size_t OFF_BOX  = 0;                               // float4[NPRED]  pred boxes
static constexpr size_t OFF_V    = OFF_BOX  + (size_t)NPRED * 16;   // float [NP1]    column potentials
static constexpr size_t OFF_MINV = OFF_V    + (size_t)NP1 * 4;      // float [NP1]
static constexpr size_t OFF_WAY  = OFF_MINV + (size_t)NP1 * 4;      // u16   [NP1]
static constexpr size_t OFF_P    = OFF_WAY  + (size_t)NP1 * 2;      // i16   [NP1]    column -> row match
static constexpr size_t OFF_USED = OFF_P    + (size_t)NP1 * 2;      // u8    [NP1]
static constexpr size_t SMEM_MATCH = (OFF_USED + (size_t)NP1 + 255) & ~(size_t)255; // ~238 KB < 320 KB

// ---------------- CDNA5 async global->LDS staging ----------------
typedef __attribute__((address_space(1))) int g_i32;   // global addrspace int
typedef __attribute__((address_space(3))) int l_i32;   // LDS addrspace int

__device__ __forceinline__ void async_g2l_b32(const float* g, float* l) {
#if __has_builtin(__builtin_amdgcn_global_load_async_to_lds_b32)
  __builtin_amdgcn_global_load_async_to_lds_b32((g_i32*)(size_t)g, (l_i32*)l, 0, 0);
#else
  unsigned loff = (unsigned)(size_t)(const void*)l;          // low 32 bits of generic shared ptr = LDS offset
  unsigned long long ga = (unsigned long long)(size_t)g;
  asm volatile("global_load_async_to_lds_b32 %0, %1, off" :: "v"(loff), "v"(ga) : "memory");
#endif
}

__device__ __forceinline__ void wait_async0() {
#if __has_builtin(__builtin_amdgcn_s_wait_asynccnt)
  __builtin_amdgcn_s_wait_asynccnt(0);
#else
  asm volatile("s_wait_asynccnt 0" ::: "memory");
#endif
}

__device__ __forceinline__ float safe_log(float x) {
  return fmaxf(logf(x), -100.0f);
}

// Deterministic block sum: wave32 shuffle tree + fixed-order partial combine (valid on tid 0).
__device__ __forceinline__ float block_sum(float x, float* red, int nwaves) {
#pragma unroll
  for (int off = 16; off > 0; off >>= 1) x += __shfl_down(x, off);
  const int tid = threadIdx.x;
  if ((tid & 31) == 0) red[tid >> 5] = x;
  __syncthreads();
  float s = 0.f;
  if (tid == 0) {
    for (int k = 0; k < nwaves; ++k) s += red[k];
  }
  __syncthreads();
  return s;
}

// =====================================================================
// Kernel A: exact Jonker-Volgenant assignment per image (mirrors _lsa on
// the transposed 30x8400 cost), cost evaluated on the fly from LDS-staged
// pred boxes. One block per image; whole solver state lives in LDS.
// =====================================================================
__global__ __launch_bounds__(MBS) void match_kernel(const float* __restrict__ preds,
                                                    const float* __restrict__ targets,
                                                    int* __restrict__ matchOut /*[B][T] pred idx per target*/) {
  extern __shared__ __align__(16) unsigned char smem[];
  float4*         box  = (float4*)(smem + OFF_BOX);
  float*          v    = (float*)(smem + OFF_V);
  float*          minv = (float*)(smem + OFF_MINV);
  unsigned short* way  = (unsigned short*)(smem + OFF_WAY);
  short*          pm   = (short*)(smem + OFF_P);
  unsigned char*  used = (unsigned char*)(smem + OFF_USED);

  __shared__ float  u_[NTGT + 1];
  __shared__ float4 tbox[NTGT];
  __shared__ float  redV[32];
  __shared__ int    redJ[32];
  __shared__ unsigned short usedList[NTGT + 2];
  __shared__ int   s_j0, s_j1, s_cnt;
  __shared__ float s_delta;

  const int tid = threadIdx.x;
  const int b   = blockIdx.x;

  // ---- stage pred boxes global->LDS via CDNA5 async copy (rows are 85 f32 => only 4B aligned) ----
  const float* pbase = preds + (size_t)b * NPRED * PDIM;
  for (int n = tid; n < NPRED; n += MBS) {
    const float* g = pbase + (size_t)n * PDIM;
    float*       l = (float*)&box[n];
    async_g2l_b32(g + 0, l + 0);
    async_g2l_b32(g + 1, l + 1);
    async_g2l_b32(g + 2, l + 2);
    async_g2l_b32(g + 3, l + 3);
  }
  for (int j = tid; j <= NPRED; j += MBS) { v[j] = 0.0f; pm[j] = 0; }
  if (tid < NTGT) {
    const float* tg = targets + ((size_t)b * NTGT + tid) * TDIM;
    tbox[tid] = make_float4(tg[1], tg[2], tg[3], tg[4]);
    u_[tid + 1] = 0.0f;
  }
  if (tid == 0) u_[0] = 0.0f;
  wait_async0();
  __syncthreads();

  for (int i = 1; i <= NTGT; ++i) {
    for (int j = tid; j <= NPRED; j += MBS) { minv[j] = 3.0e38f; used[j] = 0; }
    if (tid == 0) { pm[0] = (short)i; s_j0 = 0; s_cnt = 0; }
    __syncthreads();

    for (;;) {
      const int j0 = s_j0;
      if (tid == 0) { used[j0] = 1; usedList[s_cnt] = (unsigned short)j0; s_cnt = s_cnt + 1; }
      __syncthreads();                                  // (A)
      const int    i0  = pm[j0];
      const float  ui0 = u_[i0];
      const float4 tb  = tbox[i0 - 1];

      // scan unused columns; keep first-min (reference argmin tie-break)
      float bestv = 3.0e38f;
      int   bestj = NP1;
      for (int j = tid + 1; j <= NPRED; j += MBS) {
        if (!used[j]) {
          float4 pb  = box[j - 1];
          float  cur = fabsf(pb.x - tb.x) + fabsf(pb.y - tb.y) +
                       fabsf(pb.z - tb.z) + fabsf(pb.w - tb.w) - ui0 - v[j];
          float mv = minv[j];
          if (cur < mv) { mv = cur; minv[j] = cur; way[j] = (unsigned short)j0; }
          if (mv < bestv) { bestv = mv; bestj = j; }
        }
      }
      // wave32 (min value, min index) reduction
#pragma unroll
      for (int off = 16; off > 0; off >>= 1) {
        float ov = __shfl_down(bestv, off);
        int   oj = __shfl_down(bestj, off);
        if (ov < bestv || (ov == bestv && oj < bestj)) { bestv = ov; bestj = oj; }
      }
      if ((tid & 31) == 0) { redV[tid >> 5] = bestv; redJ[tid >> 5] = bestj; }
      __syncthreads();
      if (tid < 32) {
        bestv = redV[tid]; bestj = redJ[tid];
#pragma unroll
        for (int off = 16; off > 0; off >>= 1) {
          float ov = __shfl_down(bestv, off);
          int   oj = __shfl_down(bestj, off);
          if (ov < bestv || (ov == bestv && oj < bestj)) { bestv = ov; bestj = oj; }
        }
        if (tid == 0) { s_delta = bestv; s_j1 = bestj; }
      }
      __syncthreads();
      const float delta = s_delta;
      const int   j1    = s_j1;

      if (tid == 0) {
        for (int k = 0; k < s_cnt; ++k) {
          int jj = usedList[k];
          u_[pm[jj]] += delta;
          v[jj]      -= delta;
        }
        s_j0 = j1;
      }
      for (int j = tid + 1; j <= NPRED; j += MBS)
        if (!used[j]) minv[j] -= delta;
      __syncthreads();                                  // (B)

      if (pm[j1] == 0) {                                // reached a free column: augment
        __syncthreads();                                // all reads of pm done before t0 rewrites it
        if (tid == 0) {
          int jc = j1;
          while (jc != 0) { int jp = way[jc]; pm[jc] = pm[jp]; jc = jp; }
        }
        __syncthreads();
        break;
      }
    }
  }

  // invert: column j matched to row pm[j]  =>  target (pm[j]-1) gets pred (j-1)
  for (int j = tid + 1; j <= NPRED; j += MBS) {
    int pi = pm[j];
    if (pi > 0) matchOut[b * NTGT + (pi - 1)] = j - 1;
  }
}

// =====================================================================
// Kernel B: per-image matched losses (GIoU box, matched-obj BCE, focal cls)
// =====================================================================
__global__ __launch_bounds__(256) void matched_loss_kernel(const float* __restrict__ preds,
                                                           const float* __restrict__ targets,
                                                           const int* __restrict__ matchPred,
                                                           float* __restrict__ out3 /*[B][3]*/) {
  __shared__ int   mp[NTGT];
  __shared__ float tgt[NTGT][TDIM];
  __shared__ float red[8];
  const int tid = threadIdx.x;
  const int b   = blockIdx.x;

  if (tid < NTGT) {
    mp[tid] = matchPred[b * NTGT + tid];
    const float* tg = targets + ((size_t)b * NTGT + tid) * TDIM;
#pragma unroll
    for (int k = 0; k < TDIM; ++k) tgt[tid][k] = tg[k];
  }
  __syncthreads();

  float boxL = 0.f, objL = 0.f, clsL = 0.f;
  if (tid < NTGT) {
    const int t = tid, r = mp[t];
    const float* pp = preds + ((size_t)b * NPRED + r) * PDIM;
    __builtin_prefetch(pp + 5, 0, 0);  // warm the class row (global_prefetch_b8)
    float px = pp[0], py = pp[1], pw = pp[2], ph = pp[3];
    float tx = tgt[t][1], ty = tgt[t][2], tw = tgt[t][3], th = tgt[t][4];
    float px1 = px - pw * 0.5f, py1 = py - ph * 0.5f, px2 = px + pw * 0.5f, py2 = py + ph * 0.5f;
    float tx1 = tx - tw * 0.5f, ty1 = ty - th * 0.5f, tx2 = tx + tw * 0.5f, ty2 = ty + th * 0.5f;
    float iw    = fmaxf(fminf(px2, tx2) - fmaxf(px1, tx1), 0.f);
    float ih    = fmaxf(fminf(py2, ty2) - fmaxf(py1, ty1), 0.f);
    float inter = iw * ih;
    float pa    = (px2 - px1) * (py2 - py1);
    float ta    = (tx2 - tx1) * (ty2 - ty1);
    float uni   = pa + ta - inter;
    float iou   = inter / (uni + EPSF);
    float conv  = (fmaxf(px2, tx2) - fminf(px1, tx1)) * (fmaxf(py2, ty2) - fminf(py1, ty1));
    float giou  = iou - (conv - uni) / (conv + EPSF);
    boxL = 1.f - giou;
    objL = -safe_log(pp[4]);
  }
  for (int k = tid; k < NTGT * NCLS; k += 256) {
    int   t   = k / NCLS, c = k - t * NCLS;
    int   r   = mp[t];
    float pcl = preds[((size_t)b * NPRED + r) * PDIM + 5 + c];
    int   tc  = (int)tgt[t][0];
    float bce = (c == tc) ? -safe_log(pcl) : -safe_log(1.f - pcl);
    float pt  = expf(-bce);
    float om  = 1.f - pt;
    clsL += 0.25f * om * om * bce;   // ALPHA * (1-pt)^GAMMA * bce
  }

  float bs = block_sum(boxL, red, 8);
  float os = block_sum(objL, red, 8);
  float cs = block_sum(clsL, red, 8);
  if (tid == 0) {
    out3[b * 3 + 0] = bs;
    out3[b * 3 + 1] = os;
    out3[b * 3 + 2] = cs;
  }
}

// =====================================================================
// Kernel C: unmatched objectness BCE partial sums over B x N
// =====================================================================
#define UCHUNK 1024
#define NCHUNK ((NPRED + UCHUNK - 1) / UCHUNK)   // 9

__global__ __launch_bounds__(1024) void unmatched_obj_kernel(const float* __restrict__ preds,
                                                             const int* __restrict__ matchPred,
                                                             float* __restrict__ part /*[B][NCHUNK]*/) {
  __shared__ int   mp[NTGT];
  __shared__ float red[32];
  const int tid   = threadIdx.x;
  const int chunk = blockIdx.x;
  const int b     = blockIdx.y;
  if (tid < NTGT) mp[tid] = matchPred[b * NTGT + tid];
  __syncthreads();

  int   n = chunk * UCHUNK + tid;
  float s = 0.f;
  if (n < NPRED) {
    float obj = preds[((size_t)b * NPRED + n) * PDIM + 4];
    bool matched = false;
#pragma unroll
    for (int t = 0; t < NTGT; ++t) matched |= (mp[t] == n);
    if (!matched) s = -safe_log(1.f - obj);
  }
  s = block_sum(s, red, 32);
  if (tid == 0) part[b * NCHUNK + chunk] = s;
}

// =====================================================================
// Kernel D: deterministic fixed-order finalize -> (total, box, obj, cls)
// =====================================================================
__global__ void finalize_kernel(const float* __restrict__ p1, const float* __restrict__ p2,
                                float* __restrict__ out) {
  if (threadIdx.x == 0 && blockIdx.x == 0) {
    float box = 0.f, objm = 0.f, cls = 0.f, obju = 0.f;
    for (int b = 0; b < BDIM; ++b) {
      box  += p1[b * 3 + 0];
      objm += p1[b * 3 + 1];
      cls  += p1[b * 3 + 2];
      float s = 0.f;
      for (int c = 0; c < NCHUNK; ++c) s += p2[b * NCHUNK + c];
      obju += s / (float)(NPRED - NTGT);
    }
    float box_loss = box  / (float)(BDIM * NTGT);
    float obj_loss = objm / (float)(BDIM * NTGT) + obju / (float)BDIM;
    float cls_loss = cls  / (float)(BDIM * NTGT * NCLS);
    out[0] = box_loss + obj_loss + cls_loss;
    out[1] = box_loss;
    out[2] = obj_loss;
    out[3] = cls_loss;
  }
}

extern "C" void kernel_launch(void* const* d_in, const int* in_sizes, int n_in,
                              void* d_out, int out_size, void* d_ws, size_t ws_size,
                              hipStream_t stream) {
  const float* preds   = (const float*)d_in[0];   // [16][8400][85] f32
  const float* targets = (const float*)d_in[1];   // [16][30][5]   f32
  float*       out     = (float*)d_out;           // 4 f32: total, box, obj, cls

  char*  ws        = (char*)d_ws;
  int*   matchPred = (int*)ws;                    // B*T ints   (1920 B)
  float* p1        = (float*)(ws + 2048);         // B*3 floats
  float* p2        = (float*)(ws + 4096);         // B*NCHUNK floats

  match_kernel<<<BDIM, MBS, SMEM_MATCH, stream>>>(preds, targets, matchPred);
  matched_loss_kernel<<<BDIM, 256, 0, stream>>>(preds, targets, matchPred, p1);
  unmatched_obj_kernel<<<dim3(NCHUNK, BDIM), UCHUNK, 0, stream>>>(preds, matchPred, p2);
  finalize_kernel<<<1, 32, 0, stream>>>(p1, p2, out);
}